// ConditionedDense_10806137716754
// MI455X (gfx1250) — compile-verified
//
#include <hip/hip_runtime.h>

// ConditionedDense for MI455X (gfx1250, wave32, WMMA + async-to-LDS).
//
// out[m,o] = sum_i X[m,i] * relu( sum_p P[m,p] * Wk[p, i*32+o] )
// M = B*N = 131072, P_DIM = 64, c_in = c_out = 32.
//
// Per wave: TWO tiles of 16 positions (A-fragment reuse: 2 WMMAs per LDS
// fragment read). Hypernet tiles computed transposed,
//   D[q(16) x m(16)] = Wk^T(16x4) x P^T(4x16), V_WMMA_F32_16X16X4_F32,
// so the per-position scale X[m,i] is a pure per-lane broadcast and the
// relu/scale/accumulate epilogue is 8 VALU per tile.
// Wk slices (8 KB per i) are staged double-buffered into LDS in WMMA
// A-fragment order via GLOBAL_LOAD_ASYNC_TO_LDS_B32 (ASYNCcnt), so fragment
// reads are conflict-free ds_load_2addr_b64.

typedef __attribute__((ext_vector_type(2))) float v2f;
typedef __attribute__((ext_vector_type(4))) float v4f;
typedef __attribute__((ext_vector_type(8))) float v8f;

#define C_IN   32
#define C_OUT  32
#define P_DIM  64
#define QTOT   1024          // c_in * c_out
#define WAVES_PER_WG 8
#define WG_THREADS   256
#define TILES_PER_WAVE 2

__global__ __launch_bounds__(WG_THREADS)
void conditioned_dense_wmma(const float* __restrict__ X,
                            const float* __restrict__ P,
                            const float* __restrict__ Wk,
                            float* __restrict__ out,
                            int nbatch)
{
    // Two staging buffers: 1024 v2f (8 KB) each.
    // Element (kc, oh, lane) = { Wk[p0, q], Wk[p0+1, q] },
    //   p0 = 4*kc + 2*(lane/16),  q = i*32 + oh*16 + (lane%16).
    __shared__ v2f lds[2][1024];

    const int tid   = threadIdx.x;
    const int lane  = tid & 31;
    const int wave  = tid >> 5;
    const int lhalf = lane >> 4;      // 0 or 1
    const int l16   = lane & 15;

    // Async-stage the Wk slice for a given i into buffer b (whole workgroup).
    auto stage = [&](int b, int i) {
        const int qbase = i * 32;
        #pragma unroll
        for (int j = 0; j < 4; ++j) {
            const int e  = tid + j * WG_THREADS;   // 0..1023
            const int kc = e >> 6;
            const int oh = (e >> 5) & 1;
            const int L  = e & 31;
            const int p0 = 4 * kc + 2 * (L >> 4);
            const int q  = qbase + oh * 16 + (L & 15);
            const float* g0 = Wk + p0 * QTOT + q;
            const float* g1 = g0 + QTOT;
            // low 32 bits of a generic LDS pointer == LDS byte offset
            unsigned d0 = (unsigned)(size_t)&lds[b][e];
            asm volatile("global_load_async_to_lds_b32 %0, %1, off"
                         :: "v"(d0), "v"(g0) : "memory");
            asm volatile("global_load_async_to_lds_b32 %0, %1, off"
                         :: "v"(d0 + 4u), "v"(g1) : "memory");
        }
    };
    auto wait_async_and_barrier = [&]() {
        asm volatile("s_wait_asynccnt 0x0" ::: "memory");
        __syncthreads();
    };

    for (int batch = blockIdx.x; batch < nbatch; batch += gridDim.x) {
        const int m0 = (batch * WAVES_PER_WG + wave) * (16 * TILES_PER_WAVE);
        const int m1 = m0 + 16;

        // ---- B operands: P^T fragments for both tiles, register resident --
        // B[4 x 16]: VGPR0 = rows K0 (lanes 0-15) / K2 (lanes 16-31),
        //            VGPR1 = rows K1 / K3.  Value = P[m + l16, p].
        v2f bfrag0[16], bfrag1[16];
        {
            const float* pr0 = P + (size_t)(m0 + l16) * P_DIM + 2 * lhalf;
            const float* pr1 = P + (size_t)(m1 + l16) * P_DIM + 2 * lhalf;
            #pragma unroll
            for (int kc = 0; kc < 16; ++kc) {
                bfrag0[kc] = *(const v2f*)(pr0 + 4 * kc);
                bfrag1[kc] = *(const v2f*)(pr1 + 4 * kc);
            }
        }

        v8f acc0[2], acc1[2];
        #pragma unroll
        for (int r = 0; r < 8; ++r) {
            acc0[0][r] = 0.0f; acc0[1][r] = 0.0f;
            acc1[0][r] = 0.0f; acc1[1][r] = 0.0f;
        }

        stage(0, 0);
        wait_async_and_barrier();

        for (int i = 0; i < C_IN; ++i) {
            const v2f* cur = lds[i & 1];
            if (i + 1 < C_IN) stage((i + 1) & 1, i + 1);

            // per-lane scales X[m, i] (uniform across both lane halves)
            const float xm0 = X[(size_t)(m0 + l16) * C_IN + i];
            const float xm1 = X[(size_t)(m1 + l16) * C_IN + i];

            #pragma unroll
            for (int oh = 0; oh < 2; ++oh) {
                v8f c0 = {}, c1 = {};
                const v2f* abase = cur + oh * 32 + lane;
                #pragma unroll
                for (int kc = 0; kc < 16; ++kc) {
                    const v2f a = abase[kc * 64];     // ds_load, conflict-free
                    c0 = __builtin_amdgcn_wmma_f32_16x16x4_f32(
                            false, a, false, bfrag0[kc], (short)0, c0, false, false);
                    c1 = __builtin_amdgcn_wmma_f32_16x16x4_f32(
                            false, a, false, bfrag1[kc], (short)0, c1, false, false);
                }
                #pragma unroll
                for (int r = 0; r < 8; ++r) {
                    acc0[oh][r] += fmaxf(c0[r], 0.0f) * xm0;
                    acc1[oh][r] += fmaxf(c1[r], 0.0f) * xm1;
                }
            }
            wait_async_and_barrier();
        }

        // ---- store: element (o_local = r + 8*lhalf, m = l16) --------------
        // out[(m+l16)*32 + oh*16 + lhalf*8 + r], b128 stores
        float* op0 = out + (size_t)(m0 + l16) * C_OUT + lhalf * 8;
        float* op1 = out + (size_t)(m1 + l16) * C_OUT + lhalf * 8;
        #pragma unroll
        for (int oh = 0; oh < 2; ++oh) {
            v4f lo0, hi0, lo1, hi1;
            #pragma unroll
            for (int r = 0; r < 4; ++r) {
                lo0[r] = acc0[oh][r]; hi0[r] = acc0[oh][r + 4];
                lo1[r] = acc1[oh][r]; hi1[r] = acc1[oh][r + 4];
            }
            *(v4f*)(op0 + oh * 16)     = lo0;
            *(v4f*)(op0 + oh * 16 + 4) = hi0;
            *(v4f*)(op1 + oh * 16)     = lo1;
            *(v4f*)(op1 + oh * 16 + 4) = hi1;
        }
    }
}

extern "C" void kernel_launch(void* const* d_in, const int* in_sizes, int n_in,
                              void* d_out, int out_size, void* d_ws, size_t ws_size,
                              hipStream_t stream) {
    const float* X  = (const float*)d_in[0];
    const float* P  = (const float*)d_in[1];
    const float* Wk = (const float*)d_in[2];
    float* out = (float*)d_out;

    const int Mtot   = in_sizes[0] / C_IN;                    // B*N = 131072
    const int ntiles = Mtot / 16;                             // 8192
    const int nbatch = ntiles / (WAVES_PER_WG * TILES_PER_WAVE); // 512
    int grid = nbatch < 256 ? nbatch : 256;

    hipLaunchKernelGGL(conditioned_dense_wmma, dim3(grid), dim3(WG_THREADS), 0, stream,
                       X, P, Wk, out, nbatch);
}